// DissipationNetwork_10084583211032
// MI455X (gfx1250) — compile-verified
//
#include <hip/hip_runtime.h>
#include <stdint.h>

typedef __attribute__((ext_vector_type(2))) float v2f;
typedef __attribute__((ext_vector_type(4))) float v4f;
typedef __attribute__((ext_vector_type(4))) unsigned int u32x4;
typedef __attribute__((ext_vector_type(8))) unsigned int u32x8;

#define N_ROWS 2097152
#define NW 280  // padded weight blob size in floats (each param 4-float aligned)

// ---- packed weight blob layout (floats), every param start % 4 == 0 ----
enum : int {
  O_W_XIN = 0,    O_B_XIN = 8,    O_W_XP1 = 12,   O_B_XP1 = 28,
  O_W_XP2 = 32,   O_B_XP2 = 48,   O_W_XL1 = 52,   O_B_XL1 = 60,
  O_W_XL2 = 64,   O_B_XL2 = 80,   O_W_XL3 = 84,   O_B_XL3 = 100,
  O_W_XLO = 104,  O_B_XLO = 108,  O_W_CP1 = 112,  O_W_CP2 = 128,
  O_W_CPO = 144,  O_W_CP1M = 148, O_B_CP1M = 164, O_W_CP2M = 168,
  O_B_CP2M = 184, O_W_CPOM = 188, O_B_CPOM = 204, O_W_CLIN = 208,
  O_W_CL1 = 216,  O_W_CL2 = 224,  O_W_CLO = 232,  O_W_CLINM = 236,
  O_B_CLINM = 240, O_W_CL1M = 244, O_B_CL1M = 252, O_W_CL2M = 256,
  O_B_CL2M = 264, O_W_CLOM = 268, O_B_CLOM = 276
};

struct ParamPtrs { const float* p[35]; };

__device__ const int kOff[35] = {
  O_W_XIN, O_B_XIN, O_W_XP1, O_B_XP1, O_W_XP2, O_B_XP2,
  O_W_XL1, O_B_XL1, O_W_XL2, O_B_XL2, O_W_XL3, O_B_XL3, O_W_XLO, O_B_XLO,
  O_W_CP1, O_W_CP2, O_W_CPO,
  O_W_CP1M, O_B_CP1M, O_W_CP2M, O_B_CP2M, O_W_CPOM, O_B_CPOM,
  O_W_CLIN, O_W_CL1, O_W_CL2, O_W_CLO,
  O_W_CLINM, O_B_CLINM, O_W_CL1M, O_B_CL1M, O_W_CL2M, O_B_CL2M, O_W_CLOM, O_B_CLOM
};
__device__ const int kCnt[35] = {
  8, 4, 16, 4, 16, 4,
  8, 4, 16, 4, 16, 4, 4, 1,
  16, 16, 4,
  16, 4, 16, 4, 16, 4,
  8, 8, 8, 2,
  4, 2, 8, 2, 8, 2, 8, 2
};

// Prep: pack all 35 params into a contiguous aligned blob in d_ws, applying the
// PositiveLinear reparam (eps=1) to w_cp1/w_cp2/w_cpo once (indices 14..16),
// so the hot kernel never evaluates exp() on weights.
__global__ void prep_kernel(ParamPtrs pp, float* __restrict__ blob) {
  int t = blockIdx.x * blockDim.x + threadIdx.x;
  if (t >= NW) return;
  float v = 0.0f;
  for (int i = 0; i < 35; ++i) {
    int o = kOff[i], c = kCnt[i];
    if (t >= o && t < o + c) {
      float w = pp.p[i][t - o];
      if (i >= 14 && i <= 16)
        w = (w >= 0.0f) ? (w + __expf(-1.0f)) : __expf(w - 1.0f);
      v = w;
    }
  }
  blob[t] = v;
}

// numerically-stable softplus via hardware v_exp_f32 / v_log_f32
__device__ __forceinline__ float sp(float v) {
  return fmaxf(v, 0.0f) + __logf(1.0f + __expf(-fabsf(v)));
}
__device__ __forceinline__ v4f ld4(const float* s, int o) { return *(const v4f*)(s + o); }
__device__ __forceinline__ v2f ld2(const float* s, int o) { return *(const v2f*)(s + o); }
__device__ __forceinline__ float dot4(v4f w, v4f h) {
  return w.x * h.x + w.y * h.y + w.z * h.z + w.w * h.w;
}

// middle blocks: xs = sp(W_XL h + b + Wr_CP (xs*sp(W_CPM h + b)) + W_CL (x* ⊙ (W_CLM h + b)))
//                h  = sp(W_XP h + b)
template <int W_XL, int B_XL, int W_CPM, int B_CPM, int W_CP,
          int W_CLM, int B_CLM, int W_CL, int W_XP, int B_XP>
__device__ __forceinline__ void stage(const float* sw, v4f& xs, v4f& h, float sa, float sb) {
  v4f bcpm = ld4(sw, B_CPM);
  v4f g;
  g.x = sp(dot4(ld4(sw, W_CPM + 0), h) + bcpm.x);
  g.y = sp(dot4(ld4(sw, W_CPM + 4), h) + bcpm.y);
  g.z = sp(dot4(ld4(sw, W_CPM + 8), h) + bcpm.z);
  g.w = sp(dot4(ld4(sw, W_CPM + 12), h) + bcpm.w);
  v2f bclm = ld2(sw, B_CLM);
  float u0 = sa * (dot4(ld4(sw, W_CLM + 0), h) + bclm.x);
  float u1 = sb * (dot4(ld4(sw, W_CLM + 4), h) + bclm.y);
  v4f a = xs * g;
  v4f bxl = ld4(sw, B_XL);
  v4f c01 = ld4(sw, W_CL);      // rows 0,1 of (4x2)
  v4f c23 = ld4(sw, W_CL + 4);  // rows 2,3
  v4f t;
  t.x = dot4(ld4(sw, W_XL + 0),  h) + bxl.x + dot4(ld4(sw, W_CP + 0),  a) + c01.x * u0 + c01.y * u1;
  t.y = dot4(ld4(sw, W_XL + 4),  h) + bxl.y + dot4(ld4(sw, W_CP + 4),  a) + c01.z * u0 + c01.w * u1;
  t.z = dot4(ld4(sw, W_XL + 8),  h) + bxl.z + dot4(ld4(sw, W_CP + 8),  a) + c23.x * u0 + c23.y * u1;
  t.w = dot4(ld4(sw, W_XL + 12), h) + bxl.w + dot4(ld4(sw, W_CP + 12), a) + c23.z * u0 + c23.w * u1;
  v4f bxp = ld4(sw, B_XP);
  v4f nh;
  nh.x = sp(dot4(ld4(sw, W_XP + 0),  h) + bxp.x);
  nh.y = sp(dot4(ld4(sw, W_XP + 4),  h) + bxp.y);
  nh.z = sp(dot4(ld4(sw, W_XP + 8),  h) + bxp.z);
  nh.w = sp(dot4(ld4(sw, W_XP + 12), h) + bxp.w);
  xs.x = sp(t.x); xs.y = sp(t.y); xs.z = sp(t.z); xs.w = sp(t.w);
  h = nh;
}

__device__ __forceinline__ float row_eval(const float* sw, float xa, float xb,
                                          float sa, float sb) {
  // block 1 (from x0 directly)
  v4f clinm = ld4(sw, O_W_CLINM);
  v2f bclinm = ld2(sw, O_B_CLINM);
  float u0 = sa * (clinm.x * xa + clinm.y * xb + bclinm.x);
  float u1 = sb * (clinm.z * xa + clinm.w * xb + bclinm.y);
  v4f l1a = ld4(sw, O_W_XL1), l1b = ld4(sw, O_W_XL1 + 4);
  v4f bl1 = ld4(sw, O_B_XL1);
  v4f ca = ld4(sw, O_W_CLIN), cb = ld4(sw, O_W_CLIN + 4);
  v4f xs;
  xs.x = sp(l1a.x * xa + l1a.y * xb + bl1.x + ca.x * u0 + ca.y * u1);
  xs.y = sp(l1a.z * xa + l1a.w * xb + bl1.y + ca.z * u0 + ca.w * u1);
  xs.z = sp(l1b.x * xa + l1b.y * xb + bl1.z + cb.x * u0 + cb.y * u1);
  xs.w = sp(l1b.z * xa + l1b.w * xb + bl1.w + cb.z * u0 + cb.w * u1);
  v4f ia = ld4(sw, O_W_XIN), ib = ld4(sw, O_W_XIN + 4);
  v4f bin = ld4(sw, O_B_XIN);
  v4f h;
  h.x = sp(ia.x * xa + ia.y * xb + bin.x);
  h.y = sp(ia.z * xa + ia.w * xb + bin.y);
  h.z = sp(ib.x * xa + ib.y * xb + bin.z);
  h.w = sp(ib.z * xa + ib.w * xb + bin.w);
  // blocks 2,3
  stage<O_W_XL2, O_B_XL2, O_W_CP1M, O_B_CP1M, O_W_CP1,
        O_W_CL1M, O_B_CL1M, O_W_CL1, O_W_XP1, O_B_XP1>(sw, xs, h, sa, sb);
  stage<O_W_XL3, O_B_XL3, O_W_CP2M, O_B_CP2M, O_W_CP2,
        O_W_CL2M, O_B_CL2M, O_W_CL2, O_W_XP2, O_B_XP2>(sw, xs, h, sa, sb);
  // output block
  v4f bcpom = ld4(sw, O_B_CPOM);
  v4f g;
  g.x = sp(dot4(ld4(sw, O_W_CPOM + 0),  h) + bcpom.x);
  g.y = sp(dot4(ld4(sw, O_W_CPOM + 4),  h) + bcpom.y);
  g.z = sp(dot4(ld4(sw, O_W_CPOM + 8),  h) + bcpom.z);
  g.w = sp(dot4(ld4(sw, O_W_CPOM + 12), h) + bcpom.w);
  v2f bclom = ld2(sw, O_B_CLOM);
  float uo0 = sa * (dot4(ld4(sw, O_W_CLOM + 0), h) + bclom.x);
  float uo1 = sb * (dot4(ld4(sw, O_W_CLOM + 4), h) + bclom.y);
  v4f a = xs * g;
  v2f clo = ld2(sw, O_W_CLO);
  float t = dot4(ld4(sw, O_W_XLO), h) + sw[O_B_XLO]
          + dot4(ld4(sw, O_W_CPO), a)
          + clo.x * uo0 + clo.y * uo1;
  return sp(t);
}

// Main streaming kernel: 2 rows/thread, weights broadcast into LDS once per
// workgroup via the Tensor Data Mover (one 1120B DMA, waited on TENSORcnt).
__global__ __launch_bounds__(256) void dissnet_kernel(
    const v4f* __restrict__ x4, const v4f* __restrict__ xs4,
    const float* __restrict__ blob, v2f* __restrict__ out2) {
  __shared__ __align__(16) float sw[NW];

  if (threadIdx.x < 32) {  // wave 0 issues the TDM copy for the workgroup
    uint64_t ga = (uint64_t)(uintptr_t)blob;
    uint32_t lds = (uint32_t)(uintptr_t)(&sw[0]);  // low 32 bits = LDS byte addr
    u32x4 g0;
    g0.x = 1u;                                   // count=1, user descriptor
    g0.y = lds;                                  // lds_addr
    g0.z = (uint32_t)ga;                         // global_addr[31:0]
    g0.w = (uint32_t)((ga >> 32) & 0x1FFFFFFu) | (2u << 30);  // addr[56:32] | type=2
    u32x8 g1;
    g1.s0 = 0x00020000u;                         // data_size=4B, no flags
    g1.s1 = (NW & 0xFFFFu) << 16;                // tensor_dim0 low16 @ bits[63:48]
    g1.s2 = (NW >> 16) | (1u << 16);             // tensor_dim0 hi | tensor_dim1=1
    g1.s3 = (NW & 0xFFFFu) << 16;                // tile_dim0 @ bits[127:112]
    g1.s4 = 1u;                                  // tile_dim1=1, tile_dim2=0
    g1.s5 = NW;                                  // tensor_dim0_stride low32
    g1.s6 = 0u;
    g1.s7 = 0u;
    asm volatile("tensor_load_to_lds %0, %1" : : "s"(g0), "s"(g1) : "memory");
    __builtin_amdgcn_s_wait_tensorcnt(0);
  }
  __syncthreads();

  int i = blockIdx.x * 256 + threadIdx.x;  // each thread: rows 2i, 2i+1
  v4f xv = __builtin_nontemporal_load(x4 + i);   // x rows 2i,2i+1 (D=2)
  v4f sv = __builtin_nontemporal_load(xs4 + i);  // x_star rows 2i,2i+1
  v2f o;
  o.x = row_eval(sw, xv.x, xv.y, sv.x, sv.y);
  o.y = row_eval(sw, xv.z, xv.w, sv.z, sv.w);
  __builtin_nontemporal_store(o, out2 + i);
}

extern "C" void kernel_launch(void* const* d_in, const int* in_sizes, int n_in,
                              void* d_out, int out_size, void* d_ws, size_t ws_size,
                              hipStream_t stream) {
  (void)in_sizes; (void)n_in; (void)out_size; (void)ws_size;
  const float* x  = (const float*)d_in[0];
  const float* xs = (const float*)d_in[1];
  ParamPtrs pp;
  for (int i = 0; i < 35; ++i) pp.p[i] = (const float*)d_in[2 + i];
  float* blob = (float*)d_ws;

  prep_kernel<<<1, 288, 0, stream>>>(pp, blob);

  const int threads = N_ROWS / 2;          // 2 rows per thread
  const int blocks = threads / 256;        // 4096
  dissnet_kernel<<<blocks, 256, 0, stream>>>(
      (const v4f*)x, (const v4f*)xs, blob, (v2f*)d_out);
}